// GCN_22067541967723
// MI455X (gfx1250) — compile-verified
//
#include <hip/hip_runtime.h>

typedef float v2f __attribute__((ext_vector_type(2)));
typedef float v8f __attribute__((ext_vector_type(8)));

#define N_NODES   50000
#define N_GRAPHS  128
#define HIDDEN    128
#define FC_HID    64
#define N_CLASSES 10
#define N_EDGES   600000

// Hardware atomics (workspace is coarse-grained device memory: no remote /
// fine-grained concerns, denormal-flush behavior irrelevant for these sums).
__device__ __forceinline__ void atomAddF(float* p, float v) {
  [[clang::atomic(no_remote_memory, no_fine_grained_memory, ignore_denormal_mode)]] {
    __hip_atomic_fetch_add(p, v, __ATOMIC_RELAXED, __HIP_MEMORY_SCOPE_AGENT);
  }
}
__device__ __forceinline__ int atomAddI(int* p, int v) {
  int r;
  [[clang::atomic(no_remote_memory, no_fine_grained_memory)]] {
    r = __hip_atomic_fetch_add(p, v, __ATOMIC_RELAXED, __HIP_MEMORY_SCOPE_AGENT);
  }
  return r;
}

// ------------------------------------------------------------------
// WMMA fp32 GEMM: C[M, NT*16] = op(A[M,128]) * W[128, NT*16] (+bias)(relu)
// K fixed at 128. One wave = 16-row strip, NT 16x16 output tiles.
// Fragment layouts per CDNA5 ISA 7.12.2 (verified lowering: v_wmma_f32_16x16x4_f32).
// ------------------------------------------------------------------
template <int NT, bool RELUIN>
__global__ __launch_bounds__(256) void gemm_wmma(
    const float* __restrict__ A, const float* __restrict__ W,
    float* __restrict__ C, int M, const float* __restrict__ bias, int relu_out)
{
  const int N = NT * 16;
  const int lane  = threadIdx.x & 31;
  const int wave  = threadIdx.x >> 5;
  const int mtile = blockIdx.x * 8 + wave;
  if (mtile * 16 >= M) return;            // wave-uniform: EXEC stays all-1s for WMMA
  const int m0    = mtile * 16;
  const int row   = m0 + (lane & 15);
  const int khalf = (lane >> 4) << 1;     // 0 for lanes 0-15, 2 for lanes 16-31

  v8f acc[NT];
  #pragma unroll
  for (int t = 0; t < NT; ++t) acc[t] = {};

  const float* arow = A + (size_t)row * 128 + khalf;
  for (int k0 = 0; k0 < 128; k0 += 4) {
    v2f a = *(const v2f*)(arow + k0);
    if (RELUIN) { a[0] = fmaxf(a[0], 0.0f); a[1] = fmaxf(a[1], 0.0f); }
    const int kk = k0 + khalf;
    #pragma unroll
    for (int t = 0; t < NT; ++t) {
      const int n = t * 16 + (lane & 15);
      v2f b;
      b[0] = W[kk * N + n];
      b[1] = W[(kk + 1) * N + n];
      acc[t] = __builtin_amdgcn_wmma_f32_16x16x4_f32(
          false, a, false, b, (short)0, acc[t], false, false);
    }
  }

  const int rbase = m0 + ((lane >> 4) << 3);
  #pragma unroll
  for (int t = 0; t < NT; ++t) {
    const int c  = t * 16 + (lane & 15);
    const float bv = bias ? bias[c] : 0.0f;
    #pragma unroll
    for (int j = 0; j < 8; ++j) {
      float v = acc[t][j] + bv;
      if (relu_out) v = fmaxf(v, 0.0f);
      C[(size_t)(rbase + j) * N + c] = v;
    }
  }
}

// ------------------------------------------------------------------ CSR build
__global__ void zero_i(int* p, int n) {
  int i = blockIdx.x * blockDim.x + threadIdx.x;
  if (i < n) p[i] = 0;
}
__global__ void count_in(const int* __restrict__ dst, int* __restrict__ cnt, int nE) {
  int e = blockIdx.x * blockDim.x + threadIdx.x;
  if (e < nE) atomAddI(&cnt[dst[e]], 1);
}
// dinv = rsqrt(in_degree + 1)  (self loop included)
__global__ void dinv_kernel(const int* __restrict__ cnt, float* __restrict__ dinv) {
  int i = blockIdx.x * blockDim.x + threadIdx.x;
  if (i < N_NODES) dinv[i] = rsqrtf((float)cnt[i] + 1.0f);
}

#define SCAN_T 1024
// exclusive scan of in-degree counts -> row offsets (single block)
__global__ __launch_bounds__(SCAN_T) void scan_offsets(
    const int* __restrict__ cntin, int* __restrict__ off)
{
  __shared__ int part[SCAN_T];
  const int chunk = (N_NODES + SCAN_T - 1) / SCAN_T;   // 49
  int t  = threadIdx.x;
  int lo = t * chunk;
  int hi = lo + chunk; if (hi > N_NODES) hi = N_NODES;
  int s = 0;
  for (int i = lo; i < hi; ++i) s += cntin[i];
  part[t] = s;
  __syncthreads();
  if (t == 0) {
    int acc = 0;
    for (int i = 0; i < SCAN_T; ++i) { int v = part[i]; part[i] = acc; acc += v; }
    off[N_NODES] = acc;                                // total edge count
  }
  __syncthreads();
  int acc = part[t];
  for (int i = lo; i < hi; ++i) { off[i] = acc; acc += cntin[i]; }
}
__global__ void copy_i(const int* __restrict__ a, int* __restrict__ b, int n) {
  int i = blockIdx.x * blockDim.x + threadIdx.x;
  if (i < n) b[i] = a[i];
}
__global__ void fill_csr(const int* __restrict__ src, const int* __restrict__ dst,
                         int* __restrict__ cursor, int* __restrict__ esrc, int nE) {
  int e = blockIdx.x * blockDim.x + threadIdx.x;
  if (e < nE) {
    int p = atomAddI(&cursor[dst[e]], 1);
    esrc[p] = src[e];
  }
}

// ------------------------------------------------------------------
// Atomic-free aggregation: one wave per destination node.
// OUT[d] = bias + dinv[d] * ( dinv[d]*H[d] + sum_{s in in(d)} dinv[s]*H[s] )
// 32 lanes x float4 = full 128-float row; every row read/store is a coalesced 512B.
// ------------------------------------------------------------------
__global__ __launch_bounds__(256) void gcn_gather(
    const float* __restrict__ H, const float* __restrict__ dinv,
    const int* __restrict__ off, const int* __restrict__ esrc,
    const float* __restrict__ bias, float* __restrict__ OUT)
{
  int g = blockIdx.x * blockDim.x + threadIdx.x;       // over N_NODES*32
  if (g >= N_NODES * 32) return;
  int node = g >> 5, l = g & 31;                       // node wave-uniform
  float dn = dinv[node];
  float4 acc = *(const float4*)(H + (size_t)node * 128 + l * 4);
  acc.x *= dn; acc.y *= dn; acc.z *= dn; acc.w *= dn;  // self-loop term
  int e1 = off[node + 1];
  for (int e = off[node]; e < e1; ++e) {
    int s  = esrc[e];                                  // wave-uniform
    float w = dinv[s];
    float4 v = *(const float4*)(H + (size_t)s * 128 + l * 4);
    acc.x += w * v.x; acc.y += w * v.y; acc.z += w * v.z; acc.w += w * v.w;
  }
  float4 b4 = *(const float4*)(bias + l * 4);
  float4 o;
  o.x = b4.x + dn * acc.x; o.y = b4.y + dn * acc.y;
  o.z = b4.z + dn * acc.z; o.w = b4.w + dn * acc.w;
  *(float4*)(OUT + (size_t)node * 128 + l * 4) = o;
}

// ------------------------------------------------------------------ pooling
__global__ void zero_f(float* p, int n) {
  int i = blockIdx.x * blockDim.x + threadIdx.x;
  if (i < n) p[i] = 0.0f;
}
__global__ void pool_cnt(const int* __restrict__ batch, float* __restrict__ cnt) {
  int i = blockIdx.x * blockDim.x + threadIdx.x;
  if (i < N_NODES) atomAddF(&cnt[batch[i]], 1.0f);
}
__global__ void pool_sum(const float* __restrict__ h, const int* __restrict__ batch,
                         float* __restrict__ sums) {
  int idx = blockIdx.x * blockDim.x + threadIdx.x;     // over N_NODES*32
  if (idx >= N_NODES * 32) return;
  int i = idx >> 5, l = idx & 31;
  int g = batch[i];                                    // wave-uniform
  float4 v = *(const float4*)(h + (size_t)i * 128 + l * 4);
  float* o = sums + (size_t)g * 128 + l * 4;
  atomAddF(o + 0, fmaxf(v.x, 0.0f));                   // ReLU of layer-3 folded here
  atomAddF(o + 1, fmaxf(v.y, 0.0f));
  atomAddF(o + 2, fmaxf(v.z, 0.0f));
  atomAddF(o + 3, fmaxf(v.w, 0.0f));
}
__global__ void pool_fin(const float* __restrict__ sums, const float* __restrict__ cnt,
                         float* __restrict__ hg) {
  int idx = blockIdx.x * blockDim.x + threadIdx.x;     // over 128*128
  if (idx >= N_GRAPHS * HIDDEN) return;
  hg[idx] = sums[idx] / fmaxf(cnt[idx >> 7], 1.0f);
}

// ------------------------------------------------------------------ final FC
__global__ void fc2_kernel(const float* __restrict__ h, const float* __restrict__ W,
                           const float* __restrict__ b, float* __restrict__ out) {
  int idx = blockIdx.x * blockDim.x + threadIdx.x;     // over 128*10
  if (idx >= N_GRAPHS * N_CLASSES) return;
  int g = idx / N_CLASSES, c = idx % N_CLASSES;
  float s = b[c];
  #pragma unroll 8
  for (int k = 0; k < FC_HID; ++k) s += h[g * FC_HID + k] * W[k * N_CLASSES + c];
  out[idx] = s;
}

// ------------------------------------------------------------------
extern "C" void kernel_launch(void* const* d_in, const int* in_sizes, int n_in,
                              void* d_out, int out_size, void* d_ws, size_t ws_size,
                              hipStream_t stream) {
  const float* x   = (const float*)d_in[0];
  const int*   ei  = (const int*)d_in[1];
  const int*   src = ei;
  const int*   dst = ei + N_EDGES;
  const int*   bat = (const int*)d_in[2];
  const float* W1  = (const float*)d_in[3];  const float* b1  = (const float*)d_in[4];
  const float* W2  = (const float*)d_in[5];  const float* b2  = (const float*)d_in[6];
  const float* W3  = (const float*)d_in[7];  const float* b3  = (const float*)d_in[8];
  const float* Wf1 = (const float*)d_in[9];  const float* bf1 = (const float*)d_in[10];
  const float* Wf2 = (const float*)d_in[11]; const float* bf2 = (const float*)d_in[12];

  float* ws   = (float*)d_ws;
  float* H    = ws;                                  // [N_NODES*128]
  float* P1   = H    + (size_t)N_NODES * 128;        // [N_NODES*128]
  float* P2   = P1   + (size_t)N_NODES * 128;        // [N_NODES*128]
  float* dinv = P2   + (size_t)N_NODES * 128;        // [N_NODES]
  float* sums = dinv + N_NODES;                      // [128*128]
  float* cnt  = sums + N_GRAPHS * HIDDEN;            // [128] (contiguous after sums)
  float* hg   = cnt  + N_GRAPHS;                     // [128*128]
  float* fc1  = hg   + N_GRAPHS * HIDDEN;            // [128*64]
  int* off    = (int*)(fc1 + N_GRAPHS * FC_HID);     // [N_NODES+1]
  int* cursor = off    + (N_NODES + 1);              // [N_NODES]
  int* counts = cursor + N_NODES;                    // [N_NODES]
  int* esrc   = counts + N_NODES;                    // [N_EDGES]

  auto cdiv = [](int a, int b) { return (a + b - 1) / b; };
  const int gblk = cdiv(N_NODES / 16, 8);            // 391 blocks of 8 waves
  const int nth  = N_NODES * 32;

  // ---- one-time CSR-by-destination build (also yields dinv)
  zero_i     <<<cdiv(N_NODES, 256), 256, 0, stream>>>(counts, N_NODES);
  count_in   <<<cdiv(N_EDGES, 256), 256, 0, stream>>>(dst, counts, N_EDGES);
  dinv_kernel<<<cdiv(N_NODES, 256), 256, 0, stream>>>(counts, dinv);
  scan_offsets<<<1, SCAN_T, 0, stream>>>(counts, off);
  copy_i     <<<cdiv(N_NODES, 256), 256, 0, stream>>>(off, cursor, N_NODES);
  fill_csr   <<<cdiv(N_EDGES, 256), 256, 0, stream>>>(src, dst, cursor, esrc, N_EDGES);

  // ---- layer 1
  gemm_wmma<8, false><<<gblk, 256, 0, stream>>>(x, W1, H, N_NODES, nullptr, 0);
  gcn_gather<<<cdiv(nth, 256), 256, 0, stream>>>(H, dinv, off, esrc, b1, P1);
  // ---- layer 2 (ReLU folded into A-load)
  gemm_wmma<8, true><<<gblk, 256, 0, stream>>>(P1, W2, H, N_NODES, nullptr, 0);
  gcn_gather<<<cdiv(nth, 256), 256, 0, stream>>>(H, dinv, off, esrc, b2, P2);
  // ---- layer 3
  gemm_wmma<8, true><<<gblk, 256, 0, stream>>>(P2, W3, H, N_NODES, nullptr, 0);
  gcn_gather<<<cdiv(nth, 256), 256, 0, stream>>>(H, dinv, off, esrc, b3, P1);

  // ---- global mean pool (ReLU of layer 3 folded into pool_sum)
  zero_f  <<<cdiv(N_GRAPHS * HIDDEN + N_GRAPHS, 256), 256, 0, stream>>>(
      sums, N_GRAPHS * HIDDEN + N_GRAPHS);
  pool_cnt<<<cdiv(N_NODES, 256), 256, 0, stream>>>(bat, cnt);
  pool_sum<<<cdiv(nth, 256), 256, 0, stream>>>(P1, bat, sums);
  pool_fin<<<cdiv(N_GRAPHS * HIDDEN, 256), 256, 0, stream>>>(sums, cnt, hg);

  // ---- FC head: fc1 = relu(hg @ Wf1 + bf1) via WMMA; fc2 scalar (N=10)
  gemm_wmma<4, false><<<1, 256, 0, stream>>>(hg, Wf1, fc1, N_GRAPHS, bf1, 1);
  fc2_kernel<<<cdiv(N_GRAPHS * N_CLASSES, 256), 256, 0, stream>>>(
      fc1, Wf2, bf2, (float*)d_out);
}